// GlobalContextAttention_15985868276495
// MI455X (gfx1250) — compile-verified
//
#include <hip/hip_runtime.h>
#include <hip/hip_bf16.h>
#include <math.h>

// Shapes fixed by the reference problem.
#define C_DIM 128   // channels
#define B_SEG 128   // num_segments
#define CV4   (C_DIM / 4)   // frame stride in float4 units

typedef float v2f __attribute__((ext_vector_type(2)));
typedef float v8f __attribute__((ext_vector_type(8)));

// ---------------------------------------------------------------------------
// Kernel A: batch_index is sorted -> segment boundaries by binary search.
// seg_start[b] = first frame with idx >= b ; seg_start[B] = F.
// ---------------------------------------------------------------------------
__global__ void gca_seg_bounds(const int* __restrict__ idx, int F,
                               int* __restrict__ seg_start) {
    int b = blockIdx.x * blockDim.x + threadIdx.x;
    if (b > B_SEG) return;
    int lo = 0, hi = F;
    while (lo < hi) {
        int mid = (lo + hi) >> 1;
        if (idx[mid] < b) lo = mid + 1; else hi = mid;
    }
    seg_start[b] = lo;   // b==B_SEG naturally yields F
}

// ---------------------------------------------------------------------------
// Kernel B: per-(j,b) segment mean of x -> mean (J,B,C).
// 4 waves split the segment's frames; each lane owns 4 contiguous channels
// (one global_load_b128 per lane per frame); LDS combine across waves.
// ---------------------------------------------------------------------------
__global__ void __launch_bounds__(128)
gca_seg_mean(const float* __restrict__ x, const int* __restrict__ seg_start,
             int F, float* __restrict__ mean) {
    const int b    = blockIdx.x;
    const int j    = blockIdx.y;
    const int lane = threadIdx.x & 31;
    const int wv   = threadIdx.x >> 5;            // 0..3
    const int f0 = seg_start[b], f1 = seg_start[b + 1];

    const float4* xb = (const float4*)(x + (size_t)j * F * C_DIM);
    float4 acc = make_float4(0.f, 0.f, 0.f, 0.f);
    for (int f = f0 + wv; f < f1; f += 4) {
        const float4 v = xb[(size_t)f * CV4 + lane];   // b128, fully coalesced
        acc.x += v.x; acc.y += v.y; acc.z += v.z; acc.w += v.w;
    }

    __shared__ float lds[4 * C_DIM];
    float* row = lds + wv * C_DIM + 4 * lane;
    row[0] = acc.x; row[1] = acc.y; row[2] = acc.z; row[3] = acc.w;
    __syncthreads();

    const int c = threadIdx.x;                    // 0..127
    float s = lds[c] + lds[C_DIM + c] + lds[2 * C_DIM + c] + lds[3 * C_DIM + c];
    float cnt = (float)(f1 - f0);
    cnt = cnt < 1.f ? 1.f : cnt;                  // max(cnt, 1)
    mean[((size_t)j * B_SEG + b) * C_DIM + c] = s / cnt;
}

// ---------------------------------------------------------------------------
// Kernel C: gc = tanh(mean @ W) via V_WMMA_F32_16X16X4_F32.
// One wave (32 threads) per 16x16 output tile, K-loop of 32 WMMAs.
// Operand layouts per CDNA5 ISA 7.12.2:
//   A 16x4 : lane L -> M = L%16 ; VGPR0 = K (+2 for lanes 16-31), VGPR1 = K+1
//   B 4x16 : lane L -> N = L%16 ; VGPR0 = row K (+2 upper half), VGPR1 = K+1
//   C/D    : VGPR v -> row v (lanes 0-15) / row v+8 (lanes 16-31), col = L%16
// ---------------------------------------------------------------------------
__global__ void __launch_bounds__(32)
gca_gemm_tanh_wmma(const float* __restrict__ A,   // (M = J*B, C) row-major
                   const float* __restrict__ W,   // (C, C) row-major
                   float* __restrict__ D) {       // (M, C)
    const int m0   = blockIdx.x * 16;
    const int n0   = blockIdx.y * 16;
    const int lane = threadIdx.x;                 // 0..31, EXEC all-1s
    const int hl   = lane >> 4;                   // lane half 0/1
    const int l16  = lane & 15;

    const float* arow = A + (size_t)(m0 + l16) * C_DIM;   // A row for this lane
    const float* bcol = W + n0 + l16;                     // B column for this lane

    v8f acc = {};
    #pragma unroll 4
    for (int k = 0; k < C_DIM; k += 4) {
        v2f a, b;
        const int ka = k + 2 * hl;
        a.x = arow[ka + 0];
        a.y = arow[ka + 1];
        b.x = bcol[(size_t)(ka + 0) * C_DIM];
        b.y = bcol[(size_t)(ka + 1) * C_DIM];
        // 8 args: (neg_a, A, neg_b, B, c_mod, C, reuse_a, reuse_b)
        acc = __builtin_amdgcn_wmma_f32_16x16x4_f32(
            false, a, false, b, (short)0, acc, false, false);
    }

    #pragma unroll
    for (int v = 0; v < 8; ++v) {
        const int row = m0 + v + 8 * hl;
        D[(size_t)row * C_DIM + n0 + l16] = tanhf(acc[v]);
    }
}

// ---------------------------------------------------------------------------
// Kernel D: gate = sigmoid(<x[j,f,:], gc[j,b,:]>) ; out = segmean(gate * x).
// One block per (j,b): 8 waves, each wave handles one frame per iteration.
// Lane l owns channels 4l..4l+3 -> one global_load_b128 per lane per frame;
// dot product via wave32 shfl_xor butterfly; LDS combine across waves.
// ---------------------------------------------------------------------------
__global__ void __launch_bounds__(256)
gca_gate_scatter(const float* __restrict__ x, const float* __restrict__ gc,
                 const int* __restrict__ seg_start, int F,
                 float* __restrict__ out) {
    const int b    = blockIdx.x;
    const int j    = blockIdx.y;
    const int lane = threadIdx.x & 31;
    const int wv   = threadIdx.x >> 5;            // 0..7
    const int f0 = seg_start[b], f1 = seg_start[b + 1];

    // gc row for this segment: 4 contiguous channels per lane, in registers
    const float4 g = ((const float4*)(gc + ((size_t)j * B_SEG + b) * C_DIM))[lane];

    const float4* xb = (const float4*)(x + (size_t)j * F * C_DIM);
    float4 acc = make_float4(0.f, 0.f, 0.f, 0.f);

    for (int f = f0 + wv; f < f1; f += 8) {
        const int fn = f + 8;
        if (fn < f1)  // hint next frame into cache (global_prefetch_b8)
            __builtin_prefetch(&xb[(size_t)fn * CV4 + lane], 0, 0);

        const float4 v = xb[(size_t)f * CV4 + lane];   // b128, fully coalesced

        float d = v.x * g.x + v.y * g.y + v.z * g.z + v.w * g.w;
        #pragma unroll
        for (int m = 16; m >= 1; m >>= 1)
            d += __shfl_xor(d, m, 32);            // wave32 butterfly reduce

        const float gate = 1.f / (1.f + expf(-d));
        acc.x += gate * v.x; acc.y += gate * v.y;
        acc.z += gate * v.z; acc.w += gate * v.w;
    }

    __shared__ float lds[8 * C_DIM];
    float* row = lds + wv * C_DIM + 4 * lane;
    row[0] = acc.x; row[1] = acc.y; row[2] = acc.z; row[3] = acc.w;
    __syncthreads();

    const int c = threadIdx.x;
    if (c < C_DIM) {
        float s = 0.f;
        #pragma unroll
        for (int w = 0; w < 8; ++w) s += lds[w * C_DIM + c];
        float cnt = (float)(f1 - f0);
        cnt = cnt < 1.f ? 1.f : cnt;
        out[((size_t)j * B_SEG + b) * C_DIM + c] = s / cnt;
    }
}

// ---------------------------------------------------------------------------
// Host launcher (graph-capture safe: only kernel launches on `stream`).
// d_in order: x (J,F,C) f32 | batch_index (F,) i32 | num_segments scalar | W (C,C) f32
// ---------------------------------------------------------------------------
extern "C" void kernel_launch(void* const* d_in, const int* in_sizes, int n_in,
                              void* d_out, int out_size, void* d_ws, size_t ws_size,
                              hipStream_t stream) {
    const float* x   = (const float*)d_in[0];
    const int*   idx = (const int*)d_in[1];
    const float* W   = (const float*)d_in[3];

    const int F = in_sizes[1];
    const int J = in_sizes[0] / (F * C_DIM);
    const int M = J * B_SEG;                      // GEMM rows (multiple of 16)

    // Workspace layout: [seg_start: (B+1) ints][pad to 1KB][mean][gc]
    char*  ws        = (char*)d_ws;
    int*   seg_start = (int*)ws;
    float* mean      = (float*)(ws + 1024);
    float* gc        = mean + (size_t)M * C_DIM;
    float* out       = (float*)d_out;

    gca_seg_bounds<<<1, 256, 0, stream>>>(idx, F, seg_start);
    gca_seg_mean<<<dim3(B_SEG, J), 128, 0, stream>>>(x, seg_start, F, mean);
    gca_gemm_tanh_wmma<<<dim3(M / 16, C_DIM / 16), 32, 0, stream>>>(mean, W, gc);
    gca_gate_scatter<<<dim3(B_SEG, J), 256, 0, stream>>>(x, gc, seg_start, F, out);
}